// MLGraphConstruction_48696339202507
// MI455X (gfx1250) — compile-verified
//
#include <hip/hip_runtime.h>
#include <math.h>

typedef __attribute__((ext_vector_type(2))) float v2f;
typedef __attribute__((ext_vector_type(8))) float v8f;
typedef __attribute__((ext_vector_type(4))) unsigned int u32x4;
typedef __attribute__((ext_vector_type(8))) int i32x8;
typedef __attribute__((ext_vector_type(4))) int i32x4;

#define N_NODES 16384
#define DIM     16
#define KNN     64
#define NK      (N_NODES * KNN)
#define SLAB    512
#define CTILES  4                       // column tiles per wave in the WMMA kernel
#define MAX_RADIUS 6.0f

// ---- output offsets (in floats) ----
#define OFF_X   0                       // 16384*16      = 262144
#define OFF_EI  262144                  // 2*NK          = 2097152
#define OFF_Y   2359296                 // NK            = 1048576
#define OFF_F   3407872                 // NK*32         = 33554432
#define OFF_M   36962304                // NK            = 1048576

// ---- workspace offsets (bytes) ----
#define WS_SQ   0                       // float[16384]            = 64 KB
#define WS_NBR  65536                   // int[NK]                 = 4 MB
#define WS_D2   4259840                 // float[SLAB*16384]       = 32 MB

// toolchain arity split for the TDM builtin (see CDNA5_HIP.md)
#if defined(__has_include)
#if __has_include(<hip/amd_detail/amd_gfx1250_TDM.h>)
#define TDM_6ARG 1
#endif
#endif

// ---------------------------------------------------------------------------
// TDM: DMA one contiguous 16384-float row from global memory into LDS.
// D# layout per CDNA5 ISA:
//   group0: count=1 | lds_addr | global_addr[56:0] | type=2
//   group1: data_size=4B, tensor_dim0=16384, tensor_dim1=1,
//           tile_dim0=16384, tile_dim1=1, stride0=stride1=16384
// ---------------------------------------------------------------------------
__device__ __forceinline__ void tdm_load_row_to_lds(const float* gsrc,
                                                    unsigned lds_off) {
    unsigned long long ga = (unsigned long long)(size_t)gsrc;
    u32x4 g0;
    g0[0] = 1u;                                      // count=1 (valid user D#)
    g0[1] = lds_off;                                 // lds_addr (bytes)
    g0[2] = (unsigned)(ga & 0xFFFFFFFFu);            // global_addr[31:0]
    g0[3] = (unsigned)((ga >> 32) & 0x01FFFFFFu)     // global_addr[56:32]
          | 0x80000000u;                             // type=2 ("image")
    i32x8 g1;
    g1[0] = (int)(2u << 16);                         // data_size = 4 bytes
    g1[1] = (int)(16384u << 16);                     // tensor_dim0 low16
    g1[2] = (int)(1u << 16);                         // dim0 hi=0 | tensor_dim1=1
    g1[3] = (int)(16384u << 16);                     // dim1 hi=0 | tile_dim0=16384
    g1[4] = 1;                                       // tile_dim1=1, tile_dim2=0
    g1[5] = 16384;                                   // tensor_dim0_stride low32
    g1[6] = (int)(16384u << 16);                     // stride0 hi=0 | stride1 low16
    g1[7] = 0;
    i32x4 z4 = {0, 0, 0, 0};
#if defined(TDM_6ARG)
    i32x8 z8 = {0, 0, 0, 0, 0, 0, 0, 0};
    __builtin_amdgcn_tensor_load_to_lds(g0, g1, z4, z4, z8, 0);
#else
    __builtin_amdgcn_tensor_load_to_lds(g0, g1, z4, z4, 0);
#endif
}

// ---------------------------------------------------------------------------
// Kernel 0: row squared-norms + copy x into the output buffer
// ---------------------------------------------------------------------------
__global__ void __launch_bounds__(256)
k_sq_copy(const float* __restrict__ x, float* __restrict__ out,
          float* __restrict__ sq) {
    int i = blockIdx.x * 256 + threadIdx.x;
    if (i >= N_NODES) return;
    const float* p = x + (size_t)i * DIM;
    float s = 0.0f;
    float v[DIM];
#pragma unroll
    for (int k = 0; k < DIM; ++k) { v[k] = p[k]; s += v[k] * v[k]; }
    sq[i] = s;
    float* o = out + OFF_X + (size_t)i * DIM;
#pragma unroll
    for (int k = 0; k < DIM; ++k) o[k] = v[k];
}

// ---------------------------------------------------------------------------
// Kernel 1: d2 slab via V_WMMA_F32_16X16X4_F32 (exact f32 Gram matrix).
// One wave owns 16 query rows x (CTILES*16) candidate cols; A fragments and
// the 8 query-row norms are loaded once and reused across the column tiles.
// A (16x4 f32): lanes 0-15 hold M=lane, K={0,1}; lanes 16-31 M=lane-16, K={2,3}
// C (16x16 f32): vgpr r -> (M = r + 8*half, N = lane&15)
// ---------------------------------------------------------------------------
__global__ void __launch_bounds__(256)
k_d2_wmma(const float* __restrict__ x, const float* __restrict__ sq,
          float* __restrict__ d2, int r0) {
    const int wave = threadIdx.x >> 5;
    const int lane = threadIdx.x & 31;
    const int l16  = lane & 15;
    const int half = lane >> 4;

    const int qbase  = r0 + blockIdx.x * 16;                      // 16 query rows
    const int cbase0 = (blockIdx.y * 8 + wave) * (16 * CTILES);   // 64 cand cols

    // preload A fragments (this lane's query row, all 16 features)
    const float* ax = x + (size_t)(qbase + l16) * DIM + 2 * half;
    v2f aF[4];
#pragma unroll
    for (int j = 0; j < 4; ++j) aF[j] = (v2f){ ax[4 * j], ax[4 * j + 1] };

    // preload the 8 query-row norms this lane's accumulators need
    float sqr[8];
#pragma unroll
    for (int r = 0; r < 8; ++r) sqr[r] = sq[qbase + r + 8 * half];

#pragma unroll
    for (int ct = 0; ct < CTILES; ++ct) {
        const int ccol = cbase0 + ct * 16 + l16;
        const float* bx = x + (size_t)ccol * DIM + 2 * half;

        v8f acc = {0.f, 0.f, 0.f, 0.f, 0.f, 0.f, 0.f, 0.f};
#pragma unroll
        for (int j = 0; j < 4; ++j) {
            v2f b = { bx[4 * j], bx[4 * j + 1] };
            acc = __builtin_amdgcn_wmma_f32_16x16x4_f32(
                /*neg_a=*/false, aF[j], /*neg_b=*/false, b,
                /*c_mod=*/(short)0, acc, /*reuse_a=*/false, /*reuse_b=*/false);
        }

        const float sqc = sq[ccol];
#pragma unroll
        for (int r = 0; r < 8; ++r) {
            int   qr = qbase + r + 8 * half;
            float v  = sqr[r] + sqc - 2.0f * acc[r];
            if (qr == ccol) v = __builtin_inff();        // exclude self-loop
            d2[(size_t)(qr - r0) * N_NODES + ccol] = v;
        }
    }
}

// ---------------------------------------------------------------------------
// Kernel 2: exact top-64-smallest per row. One workgroup (256 thr) per row.
// TDM DMAs the 64 KB row into LDS (wave 0 issues, s_wait_tensorcnt 0), keys
// are converted in place to monotonic uints, then a 4-level byte radix-select
// finds the exact 64th key; candidates <= threshold are rank-sorted so the
// neighbor list is ordered by increasing distance (index tiebreak), like top_k.
// ---------------------------------------------------------------------------
__global__ void __launch_bounds__(256)
k_select(const float* __restrict__ d2, int* __restrict__ nbr, int r0) {
    __shared__ unsigned int keys[N_NODES];      // 64 KB (of 320 KB/WGP)
    __shared__ unsigned int hist[256];
    __shared__ unsigned int lkey[192];
    __shared__ int          lidx[192];
    __shared__ int          sh_kth, sh_cnt;
    __shared__ unsigned int sh_pfx;

    const int tid = threadIdx.x;
    const int row = r0 + blockIdx.x;
    const float* drow = d2 + (size_t)blockIdx.x * N_NODES;

    // --- TDM: whole row -> LDS (wave-level DMA, EXEC-independent) ---
    if (tid == 0) {
        tdm_load_row_to_lds(drow, (unsigned)(size_t)(const void*)keys);
        __builtin_amdgcn_s_wait_tensorcnt(0);
    }
    __syncthreads();

    // in-place monotonic key map: ordered uint comparison == float comparison
    for (int j = tid; j < N_NODES; j += 256) {
        unsigned int u = keys[j];
        keys[j] = (u & 0x80000000u) ? ~u : (u | 0x80000000u);
    }
    if (tid == 0) { sh_kth = KNN; sh_pfx = 0u; sh_cnt = 0; }
    __syncthreads();

    for (int level = 0; level < 4; ++level) {
        const int shift = 24 - 8 * level;
        hist[tid] = 0u;
        __syncthreads();
        unsigned int pfx = sh_pfx;
        for (int j = tid; j < N_NODES; j += 256) {
            unsigned int k = keys[j];
            bool in = (level == 0) || ((k >> (shift + 8)) == pfx);
            if (in) atomicAdd(&hist[(k >> shift) & 0xFFu], 1u);
        }
        __syncthreads();
        if (tid == 0) {
            int kth = sh_kth, cum = 0, b = 0;
            for (; b < 255; ++b) {
                int c = (int)hist[b];
                if (cum + c >= kth) break;
                cum += c;
            }
            sh_kth = kth - cum;
            sh_pfx = (sh_pfx << 8) | (unsigned int)b;
        }
        __syncthreads();
    }
    const unsigned int T = sh_pfx;   // exact 64th-smallest key

    for (int j = tid; j < N_NODES; j += 256) {
        unsigned int k = keys[j];
        if (k <= T) {
            int p = atomicAdd(&sh_cnt, 1);
            if (p < 192) { lkey[p] = k; lidx[p] = j; }
        }
    }
    __syncthreads();

    int M = sh_cnt; if (M > 192) M = 192;
    if (tid < M) {
        unsigned int kt = lkey[tid];
        int it = lidx[tid];
        int r = 0;
        for (int j = 0; j < M; ++j) {
            unsigned int kj = lkey[j];
            int ij = lidx[j];
            if (kj < kt || (kj == kt && ij < it)) ++r;
        }
        if (r < KNN) nbr[(size_t)row * KNN + r] = it;
    }
}

// ---------------------------------------------------------------------------
// Kernel 3: edge gather + features (memory bound; x is 1 MB -> L2 resident)
// ---------------------------------------------------------------------------
__global__ void __launch_bounds__(256)
k_edges(const float* __restrict__ x, const int* __restrict__ pid,
        const int* __restrict__ nbr, float* __restrict__ out) {
    int e = blockIdx.x * 256 + threadIdx.x;
    if (e >= NK) return;
    const int dst = e >> 6;
    const int src = nbr[e];

    const float* ps = x + (size_t)src * DIM;
    const float* pd = x + (size_t)dst * DIM;
    float df[DIM], sm[DIM], d2s = 0.0f;
#pragma unroll
    for (int k = 0; k < DIM; ++k) {
        float a = ps[k], b = pd[k];
        df[k] = a - b;
        sm[k] = a + b;
        d2s += df[k] * df[k];
    }
    const float dist = sqrtf(fmaxf(d2s, 0.0f));
    const bool  mk = dist < MAX_RADIUS;
    const int   pa = pid[src], pb = pid[dst];
    const bool  y  = (pa == pb) && (pa > 0) && mk;
    const float mf = mk ? 1.0f : 0.0f;

    out[OFF_EI + e]      = (float)src;
    out[OFF_EI + NK + e] = (float)dst;
    out[OFF_Y + e]       = y ? 1.0f : 0.0f;
    float* fo = out + OFF_F + (size_t)e * 32;
#pragma unroll
    for (int k = 0; k < DIM; ++k) {
        fo[k]       = df[k] * mf;
        fo[DIM + k] = sm[k] * mf;
    }
    out[OFF_M + e] = mf;
}

// ---------------------------------------------------------------------------
extern "C" void kernel_launch(void* const* d_in, const int* in_sizes, int n_in,
                              void* d_out, int out_size, void* d_ws, size_t ws_size,
                              hipStream_t stream) {
    const float* x   = (const float*)d_in[0];
    const int*   pid = (const int*)d_in[1];
    float* out = (float*)d_out;
    char*  ws  = (char*)d_ws;

    float* sq  = (float*)(ws + WS_SQ);
    int*   nbr = (int*)(ws + WS_NBR);
    float* d2  = (float*)(ws + WS_D2);

    k_sq_copy<<<N_NODES / 256, 256, 0, stream>>>(x, out, sq);

    for (int r0 = 0; r0 < N_NODES; r0 += SLAB) {
        // 32 MB d2 slab: stays resident in the 192 MB L2 between these two kernels
        k_d2_wmma<<<dim3(SLAB / 16, N_NODES / (16 * 8 * CTILES)), 256, 0, stream>>>(x, sq, d2, r0);
        k_select<<<SLAB, 256, 0, stream>>>(d2, nbr, r0);
    }

    k_edges<<<NK / 256, 256, 0, stream>>>(x, pid, nbr, out);
}